// VectorQuantizer_20040317403446
// MI455X (gfx1250) — compile-verified
//
#include <hip/hip_runtime.h>
#include <hip/hip_bf16.h>

#define NUM_EMB 1024
#define EMB_DIM 64
#define N_TOK   262144

// LDS plane layout for W: k-pair plane q (q = k/2, 0..31) holds all 1024 codes'
// float2 {w[c][2q], w[c][2q+1]} at dword offset PLANE_DW*q + 2c.
// PLANE_DW = 2080 (>= 2048, and 2080 mod 64 == 32) -> conflict-free ds_load_b64:
// bank = (32*(q+tile) mod 64) + 2r, halves of the wave on opposite bank halves.
#define PLANE_DW 2080
#define LDS_BYTES (32 * PLANE_DW * 4)   // 266240 B  (<= 320 KB/WGP)

typedef __attribute__((ext_vector_type(2))) float v2f;
typedef __attribute__((ext_vector_type(8))) float v8f;

// ---------------------------------------------------------------------------
// Kernel 1: per-code squared norms ||w_k||^2 into d_ws; zero loss accumulator.
// ---------------------------------------------------------------------------
__global__ __launch_bounds__(256) void vq_wnorm_init(const float* __restrict__ w,
                                                     float* __restrict__ wnorm,
                                                     float* __restrict__ acc) {
  int i = blockIdx.x * blockDim.x + threadIdx.x;
  if (i == 0) acc[0] = 0.0f;
  if (i < NUM_EMB) {
    const float* wr = w + (size_t)i * EMB_DIM;
    float s = 0.0f;
#pragma unroll
    for (int k = 0; k < EMB_DIM; ++k) s += wr[k] * wr[k];
    wnorm[i] = s;
  }
}

// ---------------------------------------------------------------------------
// Kernel 2: WMMA f32 distance tiles + running argmin, W staged in LDS.
// 512 threads = 16 waves. Occupancy is LDS-bound (one block / WGP, 4 waves /
// SIMD), so VGPRs are free: each wave register-tiles 2x2 -> 32 token rows x
// 32 codes per iteration = 4 independent 16-WMMA f32 accumulation chains
// sharing A and B operands. B for the whole iteration is preloaded into
// registers so the chains don't stall on individual ds loads.
//
// A layout (32-bit A 16x4): lane l<16 -> M=l, K=4t..4t+1 (v2f);
//                           lane l>=16 -> M=l-16, K=4t+2..4t+3.
// B mirrors A on K: lane needs {w[c][4t+2h], w[c][4t+2h+1]} = LDS plane q=2t+h.
// C/D layout: VGPR v, lanes 0-15: M=v, N=lane; lanes 16-31: M=v+8, N=lane-16.
// ---------------------------------------------------------------------------
__global__ __launch_bounds__(512) void vq_argmin_wmma(const float* __restrict__ z,
                                                      const float* __restrict__ w,
                                                      const float* __restrict__ wnorm,
                                                      float* __restrict__ idx_out) {
  extern __shared__ float sh[];        // 32 planes * PLANE_DW dwords

  const int tid  = threadIdx.x;
  const int lane = tid & 31;
  const int wave = tid >> 5;
  const int rowbase = (blockIdx.x * 16 + wave) * 32;   // 32 rows per wave
  const int r  = lane & 15;            // row (A) / column (B,C) within tile
  const int hh = lane >> 4;            // half-wave select
  const int h2 = hh * 2;               // K-phase offset for upper half

  // ---- Stage W into LDS planes: one code per lane -> conflict-free writes,
  //      each float2 of W read exactly once per block from L2.
  for (int i = tid; i < NUM_EMB * 32; i += 512) {
    int c = i & (NUM_EMB - 1);
    int q = i >> 10;
    *(v2f*)(sh + PLANE_DW * q + 2 * c) = *(const v2f*)(w + (size_t)c * EMB_DIM + 2 * q);
  }
  __syncthreads();

  // ---- Preload A slices for both 16-row tiles (K = 4t .. 4t+3).
  v2f a0[16], a1[16];
  const float* zrow0 = z + (size_t)(rowbase + r) * EMB_DIM + h2;
  const float* zrow1 = zrow0 + (size_t)16 * EMB_DIM;
#pragma unroll
  for (int t = 0; t < 16; ++t) {
    a0[t] = *(const v2f*)(zrow0 + 4 * t);
    a1[t] = *(const v2f*)(zrow1 + 4 * t);
  }

  float minv0[8], minv1[8];
  int   mini0[8], mini1[8];
#pragma unroll
  for (int v = 0; v < 8; ++v) {
    minv0[v] = 3.4e38f; mini0[v] = 0;
    minv1[v] = 3.4e38f; mini1[v] = 0;
  }

  for (int tp = 0; tp < NUM_EMB / 32; ++tp) {
    const int col0 = tp * 32 + r;        // code index, tile 0
    const int col1 = col0 + 16;          // code index, tile 1

    // Preload B for both code tiles (same plane, 128 B apart -> 2addr pairs).
    v2f b0[16], b1[16];
#pragma unroll
    for (int t = 0; t < 16; ++t) {
      const float* plane = sh + PLANE_DW * (2 * t + hh);
      b0[t] = *(const v2f*)(plane + 2 * col0);
      b1[t] = *(const v2f*)(plane + 2 * col1);
    }

    v8f c00 = {0.f,0.f,0.f,0.f,0.f,0.f,0.f,0.f};
    v8f c01 = {0.f,0.f,0.f,0.f,0.f,0.f,0.f,0.f};
    v8f c10 = {0.f,0.f,0.f,0.f,0.f,0.f,0.f,0.f};
    v8f c11 = {0.f,0.f,0.f,0.f,0.f,0.f,0.f,0.f};
#pragma unroll
    for (int t = 0; t < 16; ++t) {
      c00 = __builtin_amdgcn_wmma_f32_16x16x4_f32(false, a0[t], false, b0[t],
                                                  (short)0, c00, false, false);
      c01 = __builtin_amdgcn_wmma_f32_16x16x4_f32(false, a0[t], false, b1[t],
                                                  (short)0, c01, false, false);
      c10 = __builtin_amdgcn_wmma_f32_16x16x4_f32(false, a1[t], false, b0[t],
                                                  (short)0, c10, false, false);
      c11 = __builtin_amdgcn_wmma_f32_16x16x4_f32(false, a1[t], false, b1[t],
                                                  (short)0, c11, false, false);
    }

    const float wn0 = wnorm[col0];
    const float wn1 = wnorm[col1];
#pragma unroll
    for (int v = 0; v < 8; ++v) {
      // dist = ||w||^2 - 2 z.w  (||z||^2 constant per row, argmin-invariant)
      float d00 = wn0 - 2.0f * c00[v];
      float d01 = wn1 - 2.0f * c01[v];
      float d10 = wn0 - 2.0f * c10[v];
      float d11 = wn1 - 2.0f * c11[v];
      if (d00 < minv0[v]) { minv0[v] = d00; mini0[v] = col0; }
      if (d01 < minv0[v]) { minv0[v] = d01; mini0[v] = col1; }
      if (d10 < minv1[v]) { minv1[v] = d10; mini1[v] = col0; }
      if (d11 < minv1[v]) { minv1[v] = d11; mini1[v] = col1; }
    }
  }

  // ---- Cross-lane argmin within each 16-lane group (columns of one row).
#pragma unroll
  for (int v = 0; v < 8; ++v) {
    float val0 = minv0[v]; int id0 = mini0[v];
    float val1 = minv1[v]; int id1 = mini1[v];
#pragma unroll
    for (int off = 8; off >= 1; off >>= 1) {
      float o0 = __shfl_xor(val0, off, 16);
      int   i0 = __shfl_xor(id0,  off, 16);
      float o1 = __shfl_xor(val1, off, 16);
      int   i1 = __shfl_xor(id1,  off, 16);
      if (o0 < val0 || (o0 == val0 && i0 < id0)) { val0 = o0; id0 = i0; }
      if (o1 < val1 || (o1 == val1 && i1 < id1)) { val1 = o1; id1 = i1; }
    }
    if (r == 0) {
      int m = v + hh * 8;              // lanes 0-15: row v ; lanes 16-31: row v+8
      idx_out[rowbase + m]      = (float)id0;
      idx_out[rowbase + 16 + m] = (float)id1;
    }
  }
}

// ---------------------------------------------------------------------------
// Kernel 3: gather z_q = W[idx], write it, accumulate sum((z_q - z)^2).
// One float4 per thread.
// ---------------------------------------------------------------------------
__global__ __launch_bounds__(256) void vq_gather_loss(const float* __restrict__ z,
                                                      const float* __restrict__ w,
                                                      const float* __restrict__ idxf,
                                                      float* __restrict__ zq,
                                                      float* __restrict__ acc) {
  size_t e   = (size_t)blockIdx.x * blockDim.x + threadIdx.x;  // float4 index
  int    tok = (int)(e >> 4);
  int    k   = ((int)(e & 15)) << 2;
  int    idx = (int)idxf[tok];
  const float4 wq = *(const float4*)(w + (size_t)idx * EMB_DIM + k);
  const float4 zv = *(const float4*)(z + (size_t)tok * EMB_DIM + k);
  *(float4*)(zq + (size_t)tok * EMB_DIM + k) = wq;
  float dx = wq.x - zv.x, dy = wq.y - zv.y;
  float dzc = wq.z - zv.z, dwc = wq.w - zv.w;
  float s = dx * dx + dy * dy + dzc * dzc + dwc * dwc;
#pragma unroll
  for (int off = 16; off >= 1; off >>= 1) s += __shfl_down(s, off, 32);
  if ((threadIdx.x & 31) == 0) atomicAdd(acc, s);
}

__global__ void vq_finalize(const float* __restrict__ acc, float* __restrict__ loss_out) {
  // q_latent_loss == e_latent_loss == mean((z_q - z)^2) -> loss = 1.25 * mean
  loss_out[0] = 1.25f * acc[0] / (float)((size_t)N_TOK * EMB_DIM);
}

extern "C" void kernel_launch(void* const* d_in, const int* in_sizes, int n_in,
                              void* d_out, int out_size, void* d_ws, size_t ws_size,
                              hipStream_t stream) {
  const float* z = (const float*)d_in[0];   // (N_TOK, 64)
  const float* w = (const float*)d_in[1];   // (1024, 64)

  float* out  = (float*)d_out;
  float* zq   = out;                                 // N_TOK*64 floats
  float* loss = out + (size_t)N_TOK * EMB_DIM;       // 1 float
  float* idxf = loss + 1;                            // N_TOK indices (as float)

  float* wnorm = (float*)d_ws;                       // 1024 floats
  float* acc   = wnorm + NUM_EMB;                    // 1 float accumulator

  vq_wnorm_init<<<(NUM_EMB + 255) / 256, 256, 0, stream>>>(w, wnorm, acc);

  // 16 waves/block, 32 rows/wave -> 512 rows/block; W staged in 266 KB LDS.
  vq_argmin_wmma<<<N_TOK / 512, 512, LDS_BYTES, stream>>>(z, w, wnorm, idxf);

  // one float4 per thread
  vq_gather_loss<<<((size_t)N_TOK * EMB_DIM / 4) / 256, 256, 0, stream>>>(z, w, idxf, zq, acc);

  vq_finalize<<<1, 1, 0, stream>>>(acc, loss);
}